// DGCNN_Propagation_81982335746319
// MI455X (gfx1250) — compile-verified
//
#include <hip/hip_runtime.h>
#include <hip/hip_bf16.h>

typedef float v2f __attribute__((ext_vector_type(2)));
typedef float v8f __attribute__((ext_vector_type(8)));

#define KNN   16
#define NQ    2048
#define BATCH 4
#define SLOPE 0.2f
#define EPSN  1e-5f

// ---------------------------------------------------------------------------
// KNN: one thread per query; candidates staged in LDS as float4 (one
// ds_load_b128 per candidate); unrolled register top-16 insertion.
// grid: (NQ/256, B), block 256, dyn LDS = 16*G bytes
// ---------------------------------------------------------------------------
__global__ void knn_kernel(const float* __restrict__ coor_k, // (B,3,G)
                           const float* __restrict__ coor_q, // (B,3,NQ)
                           int* __restrict__ idx, int G) {
  extern __shared__ float4 s4[]; // G entries
  const int b = blockIdx.y;
  const int n = blockIdx.x * blockDim.x + threadIdx.x;
  const float* ckb = coor_k + (size_t)b * 3 * G;
  for (int i = threadIdx.x; i < G; i += blockDim.x)
    s4[i] = make_float4(ckb[i], ckb[G + i], ckb[2 * G + i], 0.f);
  __syncthreads();
  const float* cqb = coor_q + (size_t)b * 3 * NQ;
  const float qx = cqb[n], qy = cqb[NQ + n], qz = cqb[2 * NQ + n];
  float bd[KNN]; int bi[KNN];
#pragma unroll
  for (int j = 0; j < KNN; ++j) { bd[j] = 3.4e38f; bi[j] = 0; }
  for (int g = 0; g < G; ++g) {
    const float4 c = s4[g];
    const float dx = qx - c.x, dy = qy - c.y, dz = qz - c.z;
    const float d = dx * dx + dy * dy + dz * dz;
    if (d < bd[KNN - 1]) {
      bd[KNN - 1] = d; bi[KNN - 1] = g;
#pragma unroll
      for (int j = KNN - 1; j > 0; --j) {
        if (bd[j] < bd[j - 1]) {
          float td = bd[j]; bd[j] = bd[j - 1]; bd[j - 1] = td;
          int   ti = bi[j]; bi[j] = bi[j - 1]; bi[j - 1] = ti;
        }
      }
    }
  }
  int* op = idx + ((size_t)b * NQ + n) * KNN;
#pragma unroll
  for (int j = 0; j < KNN; ++j) op[j] = bi[j];
}

// ---------------------------------------------------------------------------
// Z GEMM (ctr term): Z[b,o,n] = sum_c (W[o,Chalf+c]-W[o,c]) * X[b,c,n]
// WMMA f32 16x16x4. grid: (NQ/16, O/16/TILES, B), block TILES*32.
// ---------------------------------------------------------------------------
template <int TILES>
__global__ void __launch_bounds__(TILES * 32)
zgemm_kernel(const float* __restrict__ W, int ldW, int Chalf,
             const float* __restrict__ X, int O, float* __restrict__ Z) {
  const int b    = blockIdx.z;
  const int n0   = blockIdx.x * 16;
  const int wave = threadIdx.x >> 5;
  const int lane = threadIdx.x & 31;
  const int lm   = lane & 15;
  const int klo  = (lane < 16) ? 0 : 2;
  const int o0   = (blockIdx.y * TILES + wave) * 16;
  const float* Xb = X + (size_t)b * Chalf * NQ;
  const float* Wr = W + (size_t)(o0 + lm) * ldW;
  v8f acc = {0.f, 0.f, 0.f, 0.f, 0.f, 0.f, 0.f, 0.f};
  for (int c0 = 0; c0 < Chalf; c0 += 4) {
    v2f a, bm;
    a.x = Wr[Chalf + c0 + klo]     - Wr[c0 + klo];
    a.y = Wr[Chalf + c0 + klo + 1] - Wr[c0 + klo + 1];
    bm.x = Xb[(size_t)(c0 + klo) * NQ + n0 + lm];
    bm.y = Xb[(size_t)(c0 + klo + 1) * NQ + n0 + lm];
    acc = __builtin_amdgcn_wmma_f32_16x16x4_f32(false, a, false, bm,
                                                (short)0, acc, false, false);
  }
  float* Zb = Z + (size_t)b * O * NQ;
#pragma unroll
  for (int r = 0; r < 8; ++r) {
    const int o = o0 + r + ((lane < 16) ? 0 : 8);
    Zb[(size_t)o * NQ + n0 + lm] = acc[r];
  }
}

// ---------------------------------------------------------------------------
// Gather GEMM (nbr term), TRANSPOSED tile: D[k,o] = nbr^T x Wa^T, with the
// ctr term Z[o] splatted into the WMMA C initializer. k lives in the 8 acc
// registers (+ half-wave), so max/min over k is a register tree + ONE
// shfl_xor(16); ymax/ymin stores are coalesced. Deterministic partials.
// grid: (NQ/4, B), block 256 (8 waves). Wave w owns o-tiles [w*TPW, (w+1)*TPW)
// => its channels lie entirely in group (w>>1) for both layers.
// ---------------------------------------------------------------------------
template <int TPW>
__global__ void __launch_bounds__(256)
ggemm_kernel(const float* __restrict__ W, int ldW, int Cin,
             const float* __restrict__ X, int G, const int* __restrict__ idx,
             const float* __restrict__ Z, int O,
             float* __restrict__ ymax, float* __restrict__ ymin,
             float* __restrict__ psum, float* __restrict__ psumsq) {
  const int Q    = 4;
  const int b    = blockIdx.y;
  const int n0   = blockIdx.x * Q;
  const int wave = threadIdx.x >> 5;
  const int lane = threadIdx.x & 31;
  const int lm   = lane & 15;
  const int klo  = (lane < 16) ? 0 : 2;
  const float* Xb = X + (size_t)b * Cin * G;

  // neighbor index for this lane's k-row (k = lm for A-matrix M dim)
  const int* ib = idx + ((size_t)b * NQ + n0) * KNN;
  int gq[Q];
#pragma unroll
  for (int q = 0; q < Q; ++q) gq[q] = ib[q * KNN + lm];

  // init accumulators with Z[o] (o = o0+lm, same for every k in the lane)
  v8f acc[TPW][Q];
#pragma unroll
  for (int t = 0; t < TPW; ++t) {
    const int o0 = (wave * TPW + t) * 16;
#pragma unroll
    for (int q = 0; q < Q; ++q) {
      const float zv = Z[((size_t)b * O + o0 + lm) * NQ + n0 + q];
#pragma unroll
      for (int r = 0; r < 8; ++r) acc[t][q][r] = zv;
    }
  }

  const float* Wr[TPW];
#pragma unroll
  for (int t = 0; t < TPW; ++t)
    Wr[t] = W + (size_t)((wave * TPW + t) * 16 + lm) * ldW + klo;

  for (int c0 = 0; c0 < Cin; c0 += 4) {
    v2f av[Q]; // A = nbr^T : lane k=lm, K rows c0+klo, c0+klo+1 (gathered)
#pragma unroll
    for (int q = 0; q < Q; ++q) {
      const float* p = Xb + (size_t)(c0 + klo) * G + gq[q];
      av[q].x = p[0];
      av[q].y = p[G];
    }
#pragma unroll
    for (int t = 0; t < TPW; ++t) {
      const v2f bw = *(const v2f*)(Wr[t] + c0); // B = Wa^T, 8B-aligned
#pragma unroll
      for (int q = 0; q < Q; ++q)
        acc[t][q] = __builtin_amdgcn_wmma_f32_16x16x4_f32(
            false, av[q], false, bw, (short)0, acc[t][q], false, false);
    }
  }

  // epilogue: stats + per-(o,n) max/min over k (reg tree + 1 shuffle)
  float s1 = 0.f, s2 = 0.f;
#pragma unroll
  for (int t = 0; t < TPW; ++t) {
    const int o0 = (wave * TPW + t) * 16;
#pragma unroll
    for (int q = 0; q < Q; ++q) {
      float mx = acc[t][q][0];
      float mn = acc[t][q][0];
#pragma unroll
      for (int r = 0; r < 8; ++r) {
        const float y = acc[t][q][r];
        s1 += y; s2 += y * y;
        mx = fmaxf(mx, y);
        mn = fminf(mn, y);
      }
      mx = fmaxf(mx, __shfl_xor(mx, 16, 32));
      mn = fminf(mn, __shfl_xor(mn, 16, 32));
      if (lane < 16) {
        ymax[((size_t)b * O + o0 + lm) * NQ + n0 + q] = mx;
        ymin[((size_t)b * O + o0 + lm) * NQ + n0 + q] = mn;
      }
    }
  }
#pragma unroll
  for (int d = 1; d < 32; d <<= 1) {
    s1 += __shfl_xor(s1, d, 32);
    s2 += __shfl_xor(s2, d, 32);
  }
  if (lane == 0) {
    const size_t pi = ((size_t)b * gridDim.x + blockIdx.x) * 8 + wave;
    psum[pi] = s1; psumsq[pi] = s2;
  }
}

// ---------------------------------------------------------------------------
// Reduce partials -> mu, rsqrt(var+eps) per (b,group). grid 16, block 256.
// Group g owns waves {2g, 2g+1} of every block.
// ---------------------------------------------------------------------------
__global__ void stats_kernel(const float* __restrict__ psum,
                             const float* __restrict__ psumsq, int NB,
                             float cnt, float* __restrict__ mu,
                             float* __restrict__ rs) {
  const int cell = blockIdx.x;  // b*4+g
  const int b = cell >> 2, g = cell & 3;
  __shared__ float sh1[256], sh2[256];
  float s1 = 0.f, s2 = 0.f;
  for (int i = threadIdx.x; i < NB * 2; i += blockDim.x) {
    const int nb = i >> 1, w = 2 * g + (i & 1);
    const size_t pi = ((size_t)b * NB + nb) * 8 + w;
    s1 += psum[pi]; s2 += psumsq[pi];
  }
  sh1[threadIdx.x] = s1; sh2[threadIdx.x] = s2;
  __syncthreads();
  for (int st = 128; st > 0; st >>= 1) {
    if (threadIdx.x < st) {
      sh1[threadIdx.x] += sh1[threadIdx.x + st];
      sh2[threadIdx.x] += sh2[threadIdx.x + st];
    }
    __syncthreads();
  }
  if (threadIdx.x == 0) {
    const float m = sh1[0] / cnt;
    const float v = sh2[0] / cnt - m * m;
    mu[cell] = m;
    rs[cell] = rsqrtf(v + EPSN);
  }
}

// ---------------------------------------------------------------------------
// GroupNorm affine + LeakyReLU + max-over-k (via max/min select by scale sign)
// ---------------------------------------------------------------------------
__global__ void act_kernel(const float* __restrict__ ymax,
                           const float* __restrict__ ymin,
                           const float* __restrict__ gw,
                           const float* __restrict__ gb,
                           const float* __restrict__ mu,
                           const float* __restrict__ rs, int C, int CperG,
                           float* __restrict__ out) {
  const size_t i = (size_t)blockIdx.x * blockDim.x + threadIdx.x;
  const size_t total = (size_t)BATCH * C * NQ;
  if (i >= total) return;
  const int o = (int)((i / NQ) % C);
  const int b = (int)(i / ((size_t)C * NQ));
  const int cell = b * 4 + o / CperG;
  const float scale = gw[o] * rs[cell];
  const float shift = gb[o] - mu[cell] * scale;
  const float sel = (scale >= 0.f) ? ymax[i] : ymin[i];
  const float v = sel * scale + shift;
  out[i] = (v >= 0.f) ? v : SLOPE * v;
}

// ---------------------------------------------------------------------------
extern "C" void kernel_launch(void* const* d_in, const int* in_sizes, int n_in,
                              void* d_out, int out_size, void* d_ws,
                              size_t ws_size, hipStream_t stream) {
  const float* coor  = (const float*)d_in[0];
  const float* f     = (const float*)d_in[1];
  const float* coorq = (const float*)d_in[2];
  const float* fq    = (const float*)d_in[3];
  const float* W1    = (const float*)d_in[4];
  const float* gw1   = (const float*)d_in[5];
  const float* gb1   = (const float*)d_in[6];
  const float* W2    = (const float*)d_in[7];
  const float* gw2   = (const float*)d_in[8];
  const float* gb2   = (const float*)d_in[9];

  char* ws = (char*)d_ws;
  int*   idx1 = (int*)ws;   ws += (size_t)BATCH * NQ * KNN * 4;
  int*   idx2 = (int*)ws;   ws += (size_t)BATCH * NQ * KNN * 4;
  float* h1   = (float*)ws; ws += (size_t)BATCH * 512 * NQ * 4;
  float* Zb   = (float*)ws; ws += (size_t)BATCH * 512 * NQ * 4; // reused L2
  float* ymx  = (float*)ws; ws += (size_t)BATCH * 512 * NQ * 4; // reused L2
  float* ymn  = (float*)ws; ws += (size_t)BATCH * 512 * NQ * 4; // reused L2
  float* ps   = (float*)ws; ws += (size_t)BATCH * 512 * 8 * 4;
  float* pq   = (float*)ws; ws += (size_t)BATCH * 512 * 8 * 4;
  float* mu   = (float*)ws; ws += 16 * 4;
  float* rs   = (float*)ws; ws += 16 * 4;

  // ---- Layer 1 ----
  knn_kernel<<<dim3(NQ / 256, BATCH), 256, 16 * 128, stream>>>(coor, coorq,
                                                               idx1, 128);
  zgemm_kernel<4><<<dim3(NQ / 16, 8, BATCH), 128, 0, stream>>>(W1, 768, 384,
                                                               fq, 512, Zb);
  ggemm_kernel<4><<<dim3(NQ / 4, BATCH), 256, 0, stream>>>(
      W1, 768, 384, f, 128, idx1, Zb, 512, ymx, ymn, ps, pq);
  stats_kernel<<<16, 256, 0, stream>>>(ps, pq, NQ / 4, 128.f * NQ * KNN, mu, rs);
  act_kernel<<<(BATCH * 512 * NQ + 255) / 256, 256, 0, stream>>>(
      ymx, ymn, gw1, gb1, mu, rs, 512, 128, h1);

  // ---- Layer 2 ----
  knn_kernel<<<dim3(NQ / 256, BATCH), 256, 16 * NQ, stream>>>(coorq, coorq,
                                                              idx2, NQ);
  zgemm_kernel<4><<<dim3(NQ / 16, 6, BATCH), 128, 0, stream>>>(W2, 1024, 512,
                                                               h1, 384, Zb);
  ggemm_kernel<3><<<dim3(NQ / 4, BATCH), 256, 0, stream>>>(
      W2, 1024, 512, h1, NQ, idx2, Zb, 384, ymx, ymn, ps, pq);
  stats_kernel<<<16, 256, 0, stream>>>(ps, pq, NQ / 4, 96.f * NQ * KNN, mu, rs);
  act_kernel<<<(BATCH * 384 * NQ + 255) / 256, 256, 0, stream>>>(
      ymx, ymn, gw2, gb2, mu, rs, 384, 96, (float*)d_out);
}